// CECL_90048284328353
// MI455X (gfx1250) — compile-verified
//
#include <hip/hip_runtime.h>
#include <hip/hip_bf16.h>
#include <math.h>

// ---------------------------------------------------------------------------
// MoCo-style queue update + prototype EMA for MI455X (gfx1250, wave32).
//
// Bandwidth-bound task (~400 MB of traffic -> ~17us at 23.3 TB/s). Bulk data
// movement uses float4 (b128) global loads/stores. The only real compute,
// the per-class EMA scatter, is expressed as a dense matmul
//     contrib[C,D] = W[C,N] @ q_x[N,D]   (C_pad=1008, N=256, D=128)
// and executed on the WMMA pipe with V_WMMA_F32_16X16X4_F32 (f32 in, f32 out
// -> matches the f32 reference exactly, and is deterministic unlike float
// atomics).
// ---------------------------------------------------------------------------

typedef float v2f __attribute__((ext_vector_type(2)));
typedef float v8f __attribute__((ext_vector_type(8)));

// ---------------- bulk copies ----------------

__global__ void copy_f4(const float4* __restrict__ src, float4* __restrict__ dst,
                        long long n4) {
    long long i = (long long)blockIdx.x * blockDim.x + threadIdx.x;
    long long stride = (long long)gridDim.x * blockDim.x;
    for (; i < n4; i += stride) dst[i] = src[i];
}

// new_queue = queue with rows [ptr, ptr+n) replaced by k_x
__global__ void queue_update_f4(const float4* __restrict__ queue4,
                                const float4* __restrict__ kx4,
                                float4* __restrict__ dst,
                                const int* __restrict__ ptrp,
                                int n, int d4, long long n4) {
    int ptr = *ptrp;
    long long i = (long long)blockIdx.x * blockDim.x + threadIdx.x;
    long long stride = (long long)gridDim.x * blockDim.x;
    for (; i < n4; i += stride) {
        long long row = i / d4;
        long long col = i - row * d4;
        long long off = row - ptr;
        dst[i] = (off >= 0 && off < n) ? kx4[off * d4 + col] : queue4[i];
    }
}

// targets = concat(Y, Y, queue_Y) as float
__global__ void targets_kernel(const int* __restrict__ Y,
                               const int* __restrict__ queue_Y,
                               float* __restrict__ dst,
                               int n, long long total) {
    long long i = (long long)blockIdx.x * blockDim.x + threadIdx.x;
    long long stride = (long long)gridDim.x * blockDim.x;
    for (; i < total; i += stride) {
        int v;
        if (i < n)            v = Y[i];
        else if (i < 2 * n)   v = Y[i - n];
        else                  v = queue_Y[i - 2 * n];
        dst[i] = (float)v;
    }
}

// new_queue_Y (as float) with window replaced by Y
__global__ void queueY_kernel(const int* __restrict__ queue_Y,
                              const int* __restrict__ Y,
                              float* __restrict__ dst,
                              const int* __restrict__ ptrp,
                              int n, long long qlen) {
    int ptr = *ptrp;
    long long i = (long long)blockIdx.x * blockDim.x + threadIdx.x;
    long long stride = (long long)gridDim.x * blockDim.x;
    for (; i < qlen; i += stride) {
        long long off = i - ptr;
        int v = (off >= 0 && off < n) ? Y[off] : queue_Y[i];
        dst[i] = (float)v;
    }
}

// new_isID with window set to 1.0
__global__ void isid_kernel(const float* __restrict__ isID,
                            float* __restrict__ dst,
                            const int* __restrict__ ptrp,
                            int n, long long qlen) {
    int ptr = *ptrp;
    long long i = (long long)blockIdx.x * blockDim.x + threadIdx.x;
    long long stride = (long long)gridDim.x * blockDim.x;
    for (; i < qlen; i += stride) {
        long long off = i - ptr;
        dst[i] = (off >= 0 && off < n) ? 1.0f : isID[i];
    }
}

__global__ void ptr_kernel(const int* __restrict__ ptrp, float* __restrict__ dst,
                           int n, int qlen) {
    if (threadIdx.x == 0 && blockIdx.x == 0)
        dst[0] = (float)((*ptrp + n) % qlen);
}

__global__ void zero_f(float* __restrict__ p, long long nelem) {
    long long i = (long long)blockIdx.x * blockDim.x + threadIdx.x;
    long long stride = (long long)gridDim.x * blockDim.x;
    for (; i < nelem; i += stride) p[i] = 0.0f;
}

// ---------------- EMA scatter stats (one block, deterministic) ----------------
// rank[i] = # of j<i with Y[j]==Y[i]; cnt[c] = occurrences of c in Y.
// w[i] = (1-m) * m^(cnt[Y[i]] - 1 - rank[i]); scatter W[Y[i]][i] = w[i].
// mpow[c] = m^cnt[c].
__global__ void stats_kernel(const int* __restrict__ Y,
                             float* __restrict__ W,     // [Cp][n], pre-zeroed
                             float* __restrict__ mpow,  // [Cp]
                             int n, int Cp, float m) {
    extern __shared__ int sm[];
    int* slab = sm;        // [n]
    int* scnt = sm + n;    // [Cp]
    int t = threadIdx.x;
    if (t < n) slab[t] = Y[t];
    for (int c = t; c < Cp; c += blockDim.x) scnt[c] = 0;
    __syncthreads();
    if (t < n) atomicAdd(&scnt[slab[t]], 1);   // integer atomics: deterministic
    __syncthreads();
    if (t < n) {
        int lab = slab[t];
        int rank = 0;
        for (int j = 0; j < t; ++j) rank += (slab[j] == lab) ? 1 : 0;
        float w = (1.0f - m) * powf(m, (float)(scnt[lab] - 1 - rank));
        W[(long long)lab * n + t] = w;
    }
    for (int c = t; c < Cp; c += blockDim.x)
        mpow[c] = powf(m, (float)scnt[c]);
}

// ---------------- WMMA: contrib = W @ q_x, fused EMA epilogue ----------------
// One wave (32 lanes) per 16x16 output tile. K loop in steps of 4 on
// V_WMMA_F32_16X16X4_F32.
//
// Operand layouts per ISA 7.12.2 (32-bit, 16x16x4):
//   A (16x4):  lanes 0-15 -> M=0..15 with K={0,1} in VGPR{0,1};
//              lanes 16-31 -> M=0..15 with K={2,3}.
//   B (4x16):  lanes 0-15 -> N=0..15 with K={0,1}; lanes 16-31 -> K={2,3}.
//   C/D:       VGPR r, lane l -> M = r + 8*(l>=16), N = l%16.
__global__ void wmma_proto_kernel(const float* __restrict__ W,     // [Cp][n]
                                  const float* __restrict__ feats, // [n][D]
                                  const float* __restrict__ protos,// [C][D]
                                  const float* __restrict__ mpow,  // [Cp]
                                  float* __restrict__ outP,        // [C][D]
                                  int C, int D, int n) {
    const int mt   = blockIdx.x;      // class-tile
    const int dt   = blockIdx.y;      // D-tile
    const int lane = threadIdx.x;     // 0..31, wave32
    const int half = lane >> 4;       // 0: K pair {0,1}; 1: K pair {2,3}
    const int l16  = lane & 15;

    v8f acc = {};
    const int arow = mt * 16 + l16;
    const int bcol = dt * 16 + l16;
    for (int k = 0; k < n; k += 4) {
        const int ka = k + 2 * half;
        v2f a, b;
        a.x = W[(long long)arow * n + ka];
        a.y = W[(long long)arow * n + ka + 1];
        b.x = feats[(long long)ka * D + bcol];
        b.y = feats[(long long)(ka + 1) * D + bcol];
        acc = __builtin_amdgcn_wmma_f32_16x16x4_f32(
            /*neg_a=*/false, a, /*neg_b=*/false, b,
            /*c_mod=*/(short)0, acc, /*reuse_a=*/false, /*reuse_b=*/false);
    }

    const int col = dt * 16 + l16;
#pragma unroll
    for (int r = 0; r < 8; ++r) {
        const int cls = mt * 16 + r + 8 * half;
        if (cls < C) {
            outP[(long long)cls * D + col] =
                protos[(long long)cls * D + col] * mpow[cls] + acc[r];
        }
    }
}

// ---------------- row L2-normalize (in place on d_out region) ----------------
__global__ void norm_rows_kernel(float* __restrict__ P, int D) {
    float* row = P + (long long)blockIdx.x * D;
    const int t = threadIdx.x;          // blockDim.x == D (128) -> 4 wave32s
    float v = row[t];
    float ss = v * v;
    for (int o = warpSize / 2; o > 0; o >>= 1)
        ss += __shfl_down(ss, o);
    __shared__ float sred[32];
    const int w = t / warpSize, l = t % warpSize;
    if (l == 0) sred[w] = ss;
    __syncthreads();
    if (t == 0) {
        float tot = 0.0f;
        for (int i = 0; i < (int)(blockDim.x / warpSize); ++i) tot += sred[i];
        sred[0] = tot;
    }
    __syncthreads();
    const float norm = fmaxf(sqrtf(sred[0]), 1e-12f);
    row[t] = v / norm;
}

// ---------------------------------------------------------------------------

extern "C" void kernel_launch(void* const* d_in, const int* in_sizes, int n_in,
                              void* d_out, int out_size, void* d_ws, size_t ws_size,
                              hipStream_t stream) {
    const float* q_x      = (const float*)d_in[0];
    const float* k_x      = (const float*)d_in[1];
    const float* output_x = (const float*)d_in[2];
    const float* queue    = (const float*)d_in[3];
    const float* protos   = (const float*)d_in[4];
    const float* isID     = (const float*)d_in[5];
    const int*   Y        = (const int*)d_in[6];
    const int*   queue_Y  = (const int*)d_in[7];
    const int*   ptrp     = (const int*)d_in[8];

    const int n    = in_sizes[6];          // 256
    const int qlen = in_sizes[5];          // 262144
    const int D    = in_sizes[0] / n;      // 128
    const int C    = in_sizes[2] / n;      // 1000
    const int Cp   = ((C + 15) / 16) * 16; // 1008
    const float m  = 0.99f;

    float* out = (float*)d_out;
    // output order: output_x, features, targets, new_protos, new_queue,
    //               new_queue_Y, new_isID, new_ptr
    long long off0 = 0;
    long long off1 = off0 + (long long)n * C;                 // features
    long long off2 = off1 + (long long)(2 * n + qlen) * D;    // targets
    long long off3 = off2 + (long long)(2 * n + qlen);        // new_protos
    long long off4 = off3 + (long long)C * D;                 // new_queue
    long long off5 = off4 + (long long)qlen * D;              // new_queue_Y
    long long off6 = off5 + qlen;                             // new_isID
    long long off7 = off6 + qlen;                             // new_ptr

    // workspace: W [Cp][n] + mpow [Cp]
    float* Wmat = (float*)d_ws;
    float* mpow = Wmat + (long long)Cp * n;

    const int TB = 256;
    auto blocks = [&](long long elems) {
        long long b = (elems + TB - 1) / TB;
        return (unsigned)(b > 65535 ? 65535 : b);
    };

    // --- bulk data movement (float4) ---
    long long n4_out   = (long long)n * C / 4;
    long long n4_nd    = (long long)n * D / 4;
    long long n4_queue = (long long)qlen * D / 4;
    const int d4 = D / 4;

    copy_f4<<<blocks(n4_out), TB, 0, stream>>>(
        (const float4*)output_x, (float4*)(out + off0), n4_out);
    copy_f4<<<blocks(n4_nd), TB, 0, stream>>>(
        (const float4*)q_x, (float4*)(out + off1), n4_nd);
    copy_f4<<<blocks(n4_nd), TB, 0, stream>>>(
        (const float4*)k_x, (float4*)(out + off1 + (long long)n * D), n4_nd);
    copy_f4<<<blocks(n4_queue), TB, 0, stream>>>(
        (const float4*)queue, (float4*)(out + off1 + (long long)2 * n * D), n4_queue);
    queue_update_f4<<<blocks(n4_queue), TB, 0, stream>>>(
        (const float4*)queue, (const float4*)k_x, (float4*)(out + off4),
        ptrp, n, d4, n4_queue);

    targets_kernel<<<blocks(2 * n + qlen), TB, 0, stream>>>(
        Y, queue_Y, out + off2, n, (long long)(2 * n + qlen));
    queueY_kernel<<<blocks(qlen), TB, 0, stream>>>(
        queue_Y, Y, out + off5, ptrp, n, (long long)qlen);
    isid_kernel<<<blocks(qlen), TB, 0, stream>>>(
        isID, out + off6, ptrp, n, (long long)qlen);
    ptr_kernel<<<1, 32, 0, stream>>>(ptrp, out + off7, n, qlen);

    // --- prototype EMA: zero W, stats, WMMA matmul + epilogue, normalize ---
    long long wElems = (long long)Cp * n;
    zero_f<<<blocks(wElems), TB, 0, stream>>>(Wmat, wElems);
    stats_kernel<<<1, n, (size_t)(n + Cp) * sizeof(int), stream>>>(
        Y, Wmat, mpow, n, Cp, m);

    dim3 wgrid(Cp / 16, D / 16);  // 63 x 8 tiles, one wave each
    wmma_proto_kernel<<<wgrid, 32, 0, stream>>>(
        Wmat, q_x, protos, mpow, out + off3, C, D, n);

    norm_rows_kernel<<<C, D, 0, stream>>>(out + off3, D);
}